// GHM_Loss_36155034697955
// MI455X (gfx1250) — compile-verified
//
#include <hip/hip_runtime.h>
#include <stdint.h>
#include <stddef.h>

// Problem constants (fixed by the reference)
#define B_    8
#define C_    19
#define H_    512
#define W_    512
#define N_    (H_ * W_)        // 262144 pixels per sample
#define BINS_ 10
#define NBIN_ (N_ / BINS_)     // 26214 histogram bins per sample
#define TILE_ 512              // pixels staged per block (19*512*4 = 38912 B LDS)
#define TPB_  256              // 8 waves (wave32) per block

// Workspace layout (bytes)
#define OFF_HIST 0u                  // B*NBIN u32  = 838,848 B
#define OFF_ACC  (960u * 1024u)      // 1 float accumulator
#define OFF_BETA (1024u * 1024u)     // B*NBIN f32  = 838,848 B
#define OFF_LOGP (2048u * 1024u)     // B*N f32     = 8,388,608 B

// Pass 1 (defined first so the disasm snippet shows the async-to-LDS path):
// async-stage a [C=19, TILE=512] tile of logits into LDS, compute log-softmax
// at the target class, histogram the gradient magnitude.
__global__ __launch_bounds__(TPB_) void k_pass1(const float* __restrict__ x,
                                                const int* __restrict__ tgt,
                                                unsigned* __restrict__ hist,
                                                float* __restrict__ logp) {
    __shared__ float sh[C_ * TILE_];   // only LDS object -> starts at LDS offset 0

    const int t    = threadIdx.x;
    const int b    = blockIdx.x / (N_ / TILE_);
    const int tile = blockIdx.x % (N_ / TILE_);
    const int base = tile * TILE_;

    // ---- CDNA5 async global->LDS staging (ASYNCcnt path) ----
    // Each thread issues 19 x B64 async loads: lane t covers floats [2t, 2t+1]
    // of each 512-float class row. 256 lanes * 8B = 2048B = one full row.
    // NT hint: x is a 152MB one-shot stream; keep the hot ~10MB working set
    // (histogram / logp / beta) resident in the 192MB L2 instead.
    {
        const float* g0 = x + ((size_t)b * C_) * (size_t)N_ + (size_t)(base + 2 * t);
        const unsigned lo = (unsigned)(8 * t);   // byte offset of this lane in a row
#pragma unroll
        for (int c = 0; c < C_; ++c) {
            const float* ga = g0 + (size_t)c * (size_t)N_;
            unsigned la = lo + (unsigned)c * (TILE_ * 4u);  // LDS byte address
            asm volatile("global_load_async_to_lds_b64 %0, %1, off th:TH_LOAD_NT"
                         :: "v"(la), "v"(ga)
                         : "memory");
        }
    }
    asm volatile("s_wait_asynccnt 0x0" ::: "memory");  // our async writes landed
    __syncthreads();                                   // everyone else's too

    // ---- per-pixel log-softmax + histogram ----
#pragma unroll
    for (int k = 0; k < TILE_ / TPB_; ++k) {
        const int p  = t + k * TPB_;
        const int gp = base + p;

        float vals[C_];
        float m = -3.402823466e38f;
#pragma unroll
        for (int c = 0; c < C_; ++c) {
            vals[c] = sh[c * TILE_ + p];
            m = fmaxf(m, vals[c]);
        }
        float s = 0.0f;
#pragma unroll
        for (int c = 0; c < C_; ++c) s += __expf(vals[c] - m);

        const int tc = tgt[(size_t)b * N_ + gp];
        const float lp = sh[tc * TILE_ + p] - m - __logf(s);
        logp[(size_t)b * N_ + gp] = lp;

        const float g = 1.0f - __expf(lp);   // |p_t - 1|, p_t <= 1
        int bin = (int)floorf(g * ((float)NBIN_ - 0.0001f));
        bin = bin < 0 ? 0 : (bin > NBIN_ - 1 ? NBIN_ - 1 : bin);
        atomicAdd(&hist[b * NBIN_ + bin], 1u);
    }
}

__global__ __launch_bounds__(256) void k_init(unsigned* __restrict__ hist,
                                              float* __restrict__ acc) {
    int i = blockIdx.x * blockDim.x + threadIdx.x;
    if (i < B_ * NBIN_) hist[i] = 0u;
    if (i == 0) *acc = 0.0f;
}

// Pass 2: one block per batch sample: nonempty-bin count, then beta = N / gd.
__global__ __launch_bounds__(256) void k_beta(const unsigned* __restrict__ hist,
                                              float* __restrict__ beta) {
    __shared__ unsigned red[256];
    const int b = blockIdx.x, t = threadIdx.x;

    unsigned cnt = 0;
    for (int i = t; i < NBIN_; i += 256) cnt += (hist[b * NBIN_ + i] != 0u) ? 1u : 0u;
    red[t] = cnt;
    __syncthreads();
    for (int s = 128; s > 0; s >>= 1) {
        if (t < s) red[t] += red[t + s];
        __syncthreads();
    }
    const float ne = (float)red[0];

    for (int i = t; i < NBIN_; i += 256) {
        float bc = (float)hist[b * NBIN_ + i] * ne;   // bin_count * nonempty
        float gd = bc > 1.0f ? bc : 1.0f;             // clamp(min=1)
        beta[b * NBIN_ + i] = (float)N_ / gd;
    }
}

// Pass 3: weighted NLL accumulation. Recompute the bin from stored logp_t with
// bit-identical float ops -> same bin as pass 1.
__global__ __launch_bounds__(256) void k_loss(const float* __restrict__ logp,
                                              const float* __restrict__ beta,
                                              float* __restrict__ acc) {
    __shared__ float red[256];
    const int t = threadIdx.x;

    float sum = 0.0f;
    const long total = (long)B_ * (long)N_;
    for (long i = (long)blockIdx.x * blockDim.x + t; i < total;
         i += (long)gridDim.x * blockDim.x) {
        const int b = (int)(i / N_);       // N_ = 2^18 -> shift
        const float lp = logp[i];
        const float g = 1.0f - __expf(lp);
        int bin = (int)floorf(g * ((float)NBIN_ - 0.0001f));
        bin = bin < 0 ? 0 : (bin > NBIN_ - 1 ? NBIN_ - 1 : bin);
        sum += beta[b * NBIN_ + bin] * (-lp);
    }
    red[t] = sum;
    __syncthreads();
    for (int s = 128; s > 0; s >>= 1) {
        if (t < s) red[t] += red[t + s];
        __syncthreads();
    }
    if (t == 0) atomicAdd(acc, red[0]);
}

__global__ void k_fin(const float* __restrict__ acc, float* __restrict__ out) {
    if (threadIdx.x == 0 && blockIdx.x == 0)
        out[0] = acc[0] * (1.0f / ((float)B_ * (float)N_));
}

extern "C" void kernel_launch(void* const* d_in, const int* in_sizes, int n_in,
                              void* d_out, int out_size, void* d_ws, size_t ws_size,
                              hipStream_t stream) {
    (void)in_sizes; (void)n_in; (void)out_size; (void)ws_size;

    const float* x   = (const float*)d_in[0];   // [B,C,H,W] f32
    const int*   tgt = (const int*)d_in[1];     // [B,H,W]   i32

    char* ws = (char*)d_ws;
    unsigned* hist = (unsigned*)(ws + OFF_HIST);
    float*    acc  = (float*)(ws + OFF_ACC);
    float*    beta = (float*)(ws + OFF_BETA);
    float*    logp = (float*)(ws + OFF_LOGP);
    float*    out  = (float*)d_out;

    k_init <<<(B_ * NBIN_ + 255) / 256, 256, 0, stream>>>(hist, acc);
    k_pass1<<<B_ * (N_ / TILE_),        TPB_, 0, stream>>>(x, tgt, hist, logp);
    k_beta <<<B_,                       256,  0, stream>>>(hist, beta);
    k_loss <<<2048,                     256,  0, stream>>>(logp, beta, acc);
    k_fin  <<<1,                        64,   0, stream>>>(acc, out);
}